// RNN_81604378624001
// MI455X (gfx1250) — compile-verified
//
#include <hip/hip_runtime.h>
#include <hip/hip_bf16.h>

typedef __attribute__((ext_vector_type(2))) float v2f;
typedef __attribute__((ext_vector_type(4))) float v4f;
typedef __attribute__((ext_vector_type(8))) float v8f;

// Half-wave broadcast-swap (lane i <-> lane i^16) as a pure-VALU op.
// Used only in the epilogue reduction.
__device__ __forceinline__ float xswap16(float v) {
#if __has_builtin(__builtin_amdgcn_permlanex16)
    return __builtin_bit_cast(float, __builtin_amdgcn_permlanex16(
        __builtin_bit_cast(int, v), __builtin_bit_cast(int, v),
        0x76543210, 0xfedcba98, false, false));
#elif __has_builtin(__builtin_amdgcn_ds_swizzle)
    return __builtin_bit_cast(float,
        __builtin_amdgcn_ds_swizzle(__builtin_bit_cast(int, v), 0x401f));
#else
    return v; // host pass only; never executed
#endif
}

// Single-instruction ReLU: v_med3_f32(x, 0, +inf). Inline asm so AMDGPU
// instcombine can't fold it back into canonicalize+maxnum (2 ops).
__device__ __forceinline__ float relu_med3(float x, float finf) {
#if defined(__AMDGCN__)
    float r;
    asm("v_med3_f32 %0, %1, 0, %2" : "=v"(r) : "v"(x), "v"(finf));
    return r;
#else
    return fmaxf(x, 0.0f);
#endif
}

// ReLU-RNN scan, one wave handles 2 tiles of 16 batch rows.
//
// Layout trick: the f32 16x16 C/D operand holds H[v + 8*hi][n] in register v
// at lane-half hi. Feeding D straight back as the B operand (chunk j =
// registers (d[2j], d[2j+1])) presents K-rows in the fixed permuted order
//   r(j,hi,e) = 2j + e + 8*hi          (a bijection over K = 0..15)
// instead of the canonical 4j + 2hi + e. Since A and B contract over K, we
// absorb this permutation into the A operand ONCE at load time:
//   a[j].{x,y} = W_hh[m][2j + 8*hi + {0,1}]
// so NO cross-lane data movement is needed between timesteps at all.
__global__ __launch_bounds__(256) void rnn_scan_wmma(
    const float* __restrict__ x,     // [B, T] (I == 1)
    const float* __restrict__ W_ih,  // [16, 1]
    const float* __restrict__ b_ih,  // [16]
    const float* __restrict__ W_hh,  // [16, 16] row-major
    const float* __restrict__ b_hh,  // [16]
    const float* __restrict__ W_fc,  // [2, 16]
    const float* __restrict__ b_fc,  // [2]
    float* __restrict__ out,         // [B, 2]
    int B, int T)
{
    const int lane = threadIdx.x & 31;
    const int wave = threadIdx.x >> 5;
    const int n    = lane & 15;          // column (batch within tile)
    const bool hi  = lane >= 16;
    const int h8   = hi ? 8 : 0;

    const int waveGlobal = blockIdx.x * (blockDim.x >> 5) + wave;
    const int row0 = (waveGlobal * 2 + 0) * 16;   // first batch row of tile 0
    const int row1 = (waveGlobal * 2 + 1) * 16;   // first batch row of tile 1
    if (row0 >= B) return;                        // whole-wave uniform exit

    const float finf = __builtin_bit_cast(float, 0x7f800000); // +inf

    // W_hh in PERMUTED A-operand layout (see comment above):
    // a[j].x = W_hh[m][2j + 8hi], a[j].y = W_hh[m][2j + 8hi + 1], m = n.
    v2f a[4];
#pragma unroll
    for (int j = 0; j < 4; ++j) {
        const int k0 = 2 * j + h8;
        a[j].x = W_hh[n * 16 + k0];
        a[j].y = W_hh[n * 16 + k0 + 1];
    }

    // Per-lane slices of W_ih and combined bias (b_ih + b_hh) in C/D layout rows.
    float wih[8], bias[8];
#pragma unroll
    for (int v = 0; v < 8; ++v) {
        wih[v]  = W_ih[v + h8];
        bias[v] = b_ih[v + h8] + b_hh[v + h8];
    }

    // Hidden state, kept in C/D register layout (doubles as the B operand).
    v8f h0 = {}, h1 = {};
    const v8f zc = {};   // zero C for the second partial-accumulation chain

    const float* xr0 = x + (size_t)(row0 + n) * T;
    const float* xr1 = x + (size_t)(row1 + n) * T;

    for (int t = 0; t < T; t += 4) {
        v4f xq0 = *(const v4f*)(xr0 + t);
        v4f xq1 = *(const v4f*)(xr1 + t);
        __builtin_prefetch(xr0 + t + 32, 0, 1);
        __builtin_prefetch(xr1 + t + 32, 0, 1);
#pragma unroll
        for (int u = 0; u < 4; ++u) {
            // C := xi_t = x[b,t] * W_ih + (b_ih + b_hh)  (off the serial chain)
            v8f c0, c1;
#pragma unroll
            for (int v = 0; v < 8; ++v) {
                c0[v] = fmaf(xq0[u], wih[v], bias[v]);
                c1[v] = fmaf(xq1[u], wih[v], bias[v]);
            }
            // B chunks are just even register pairs of the hidden state.
            v2f b00 = { h0[0], h0[1] }, b01 = { h0[2], h0[3] };
            v2f b02 = { h0[4], h0[5] }, b03 = { h0[6], h0[7] };
            v2f b10 = { h1[0], h1[1] }, b11 = { h1[2], h1[3] };
            v2f b12 = { h1[4], h1[5] }, b13 = { h1[6], h1[7] };

            // K=16 split into TWO independent 2-deep accumulation chains per
            // tile (chain depth 2 WMMAs + 1 add instead of 4 WMMAs), with the
            // two batch tiles interleaved for additional ILP.
            v8f q0 = __builtin_amdgcn_wmma_f32_16x16x4_f32(
                        false, a[2], false, b02, (short)0, zc, false, false);
            v8f q1 = __builtin_amdgcn_wmma_f32_16x16x4_f32(
                        false, a[2], false, b12, (short)0, zc, false, false);
            c0 = __builtin_amdgcn_wmma_f32_16x16x4_f32(
                        false, a[0], false, b00, (short)0, c0, false, false);
            c1 = __builtin_amdgcn_wmma_f32_16x16x4_f32(
                        false, a[0], false, b10, (short)0, c1, false, false);
            q0 = __builtin_amdgcn_wmma_f32_16x16x4_f32(
                        false, a[3], false, b03, (short)0, q0, false, false);
            q1 = __builtin_amdgcn_wmma_f32_16x16x4_f32(
                        false, a[3], false, b13, (short)0, q1, false, false);
            c0 = __builtin_amdgcn_wmma_f32_16x16x4_f32(
                        false, a[1], false, b01, (short)0, c0, false, false);
            c1 = __builtin_amdgcn_wmma_f32_16x16x4_f32(
                        false, a[1], false, b11, (short)0, c1, false, false);

            // h = relu(partial0 + partial1): v_pk_add + single v_med3 each.
#pragma unroll
            for (int v = 0; v < 8; ++v) {
                h0[v] = relu_med3(c0[v] + q0[v], finf);
                h1[v] = relu_med3(c1[v] + q1[v], finf);
            }
        }
    }

    // Epilogue: out[b, c] = sum_h hT[b,h] * W_fc[c,h] + b_fc[c].
    // h holds hT in C/D layout: lane n has rows 0..7, lane n+16 rows 8..15.
    float wf0[8], wf1[8];
#pragma unroll
    for (int v = 0; v < 8; ++v) {
        wf0[v] = W_fc[0 * 16 + v + h8];
        wf1[v] = W_fc[1 * 16 + v + h8];
    }
    float p00 = 0.f, p01 = 0.f, p10 = 0.f, p11 = 0.f;
#pragma unroll
    for (int v = 0; v < 8; ++v) {
        p00 = fmaf(h0[v], wf0[v], p00);
        p01 = fmaf(h0[v], wf1[v], p01);
        p10 = fmaf(h1[v], wf0[v], p10);
        p11 = fmaf(h1[v], wf1[v], p11);
    }
    // Cross-half reduction (h=0..7 partial + h=8..15 partial).
    p00 += xswap16(p00); p01 += xswap16(p01);
    p10 += xswap16(p10); p11 += xswap16(p11);

    if (!hi) {
        out[(size_t)(row0 + n) * 2 + 0] = p00 + b_fc[0];
        out[(size_t)(row0 + n) * 2 + 1] = p01 + b_fc[1];
        if (row1 < B) {
            out[(size_t)(row1 + n) * 2 + 0] = p10 + b_fc[0];
            out[(size_t)(row1 + n) * 2 + 1] = p11 + b_fc[1];
        }
    }
}

extern "C" void kernel_launch(void* const* d_in, const int* in_sizes, int n_in,
                              void* d_out, int out_size, void* d_ws, size_t ws_size,
                              hipStream_t stream) {
    const float* x    = (const float*)d_in[0];
    const float* W_ih = (const float*)d_in[1];
    const float* b_ih = (const float*)d_in[2];
    const float* W_hh = (const float*)d_in[3];
    const float* b_hh = (const float*)d_in[4];
    const float* W_fc = (const float*)d_in[5];
    const float* b_fc = (const float*)d_in[6];
    float* out = (float*)d_out;

    const int C = 2;
    const int B = out_size / C;          // 4096
    const int T = in_sizes[0] / B;       // I == 1 -> 2048

    const int threads = 256;                              // 8 waves/block
    const int batchPerBlock = (threads / 32) * 2 * 16;    // 2 tiles of 16 per wave = 256
    const int blocks = (B + batchPerBlock - 1) / batchPerBlock;

    rnn_scan_wmma<<<blocks, threads, 0, stream>>>(
        x, W_ih, b_ih, W_hh, b_hh, W_fc, b_fc, out, B, T);
}